// RandomMaskSubgraphs_1683627180465
// MI455X (gfx1250) — compile-verified
//
#include <hip/hip_runtime.h>
#include <stdint.h>

// ---------------------------------------------------------------------------
// RandomMaskSubgraphs for MI455X (gfx1250).
// Bandwidth-bound edge-stream pipeline:
//   - edge stream (rows/cols) staged into LDS via CDNA5 async-to-LDS
//     (global_load_async_to_lds_b128 + s_wait_asynccnt), double buffered.
//   - node-state arrays (160KB each) are L2/WGP$ resident gathers.
//   - degree histogram via global_atomic_add_f32 (L2 atomics).
//   - deterministic wave32 ballot-based compaction of masked nodes.
// No matrix math in this workload -> no WMMA by design.
// ---------------------------------------------------------------------------

#define TILE_EDGES 1024   // edges per tile per block (256 threads x 4 via b128)

__device__ __forceinline__ void wait_async0() {
    asm volatile("s_wait_asynccnt 0x0" ::: "memory");
}

// Issue async copy of one tile (rows+cols) into LDS buffer b.
// Full tiles use the CDNA5 async DMA path; a (never-taken for E%1024==0)
// tail falls back to guarded direct LDS stores.
__device__ __forceinline__ void issue_tile(const int* __restrict__ rows,
                                           const int* __restrict__ cols,
                                           long long base, long long E,
                                           int (*smr)[TILE_EDGES],
                                           int (*smc)[TILE_EDGES],
                                           int b, int t) {
    if (base + TILE_EDGES <= E) {
        uint32_t lr = (uint32_t)(uintptr_t)&smr[b][t * 4];
        uint32_t lc = (uint32_t)(uintptr_t)&smc[b][t * 4];
        const int* gr = rows + base + t * 4;
        const int* gc = cols + base + t * 4;
        asm volatile("global_load_async_to_lds_b128 %0, %1, off"
                     :: "v"(lr), "v"(gr) : "memory");
        asm volatile("global_load_async_to_lds_b128 %0, %1, off"
                     :: "v"(lc), "v"(gc) : "memory");
    } else {
        for (int k = 0; k < 4; ++k) {
            long long e = base + t + k * 256;
            smr[b][t + k * 256] = (e < E) ? rows[e] : 0;
            smc[b][t + k * 256] = (e < E) ? cols[e] : 0;
        }
    }
}

__device__ __forceinline__ unsigned int hashu(unsigned int x) {
    x ^= x >> 16; x *= 0x7feb352dU;
    x ^= x >> 15; x *= 0x846ca68bU;
    x ^= x >> 16; return x;
}

// ------------------------------ init kernels -------------------------------

__global__ void init_nodes_kernel(int* cur, int* nxt, int* msk, int* midx,
                                  float* deg, float* dinv, int n) {
    int i = blockIdx.x * blockDim.x + threadIdx.x;
    if (i < n) {
        cur[i] = 0; nxt[i] = 0; msk[i] = 0; midx[i] = 0;
        deg[i] = 0.0f; dinv[i] = 0.0f;
    }
}

__global__ void init_edges_kernel(int* alive, long long E) {
    long long stride = (long long)gridDim.x * blockDim.x;
    for (long long e = blockIdx.x * (long long)blockDim.x + threadIdx.x;
         e < E; e += stride)
        alive[e] = 1;
}

__global__ void zero_out_kernel(float* out, long long n) {
    long long stride = (long long)gridDim.x * blockDim.x;
    for (long long i = blockIdx.x * (long long)blockDim.x + threadIdx.x;
         i < n; i += stride)
        out[i] = 0.0f;
}

__global__ void scatter_seeds_kernel(const int* __restrict__ seeds, int S,
                                     int* cur, int* msk, int n) {
    int i = blockIdx.x * blockDim.x + threadIdx.x;
    if (i < S) {
        int s = seeds[i];
        if (s >= 0 && s < n) { cur[s] = 1; msk[s] = 1; }
    }
}

__global__ void keep_sample_kernel(int* msk, int n, int samp) {
    int i = blockIdx.x * blockDim.x + threadIdx.x;
    if (i < samp) {
        unsigned h = hashu(0x12345u ^ (unsigned)i);
        msk[h % (unsigned)n] = 1;
    }
}

// --------------------------- edge-stream kernels ---------------------------

// One BFS level: kill alive edges touching cur; if !last, endpoints of killed
// edges become next frontier and join the mask set.
__global__ void bfs_level_kernel(const int* __restrict__ rows,
                                 const int* __restrict__ cols,
                                 int* __restrict__ alive,
                                 const int* __restrict__ cur,
                                 int* __restrict__ nxt,
                                 int* __restrict__ msk,
                                 long long E, int last) {
    __shared__ int smr[2][TILE_EDGES];
    __shared__ int smc[2][TILE_EDGES];
    int t = threadIdx.x;
    long long numTiles = (E + TILE_EDGES - 1) / TILE_EDGES;
    long long stride = gridDim.x;
    long long tile = blockIdx.x;
    int b = 0;
    if (tile < numTiles) issue_tile(rows, cols, tile * TILE_EDGES, E, smr, smc, b, t);
    for (; tile < numTiles; tile += stride) {
        wait_async0();
        __syncthreads();
        long long nt = tile + stride;
        if (nt < numTiles) issue_tile(rows, cols, nt * TILE_EDGES, E, smr, smc, b ^ 1, t);
        long long base = tile * TILE_EDGES;
#pragma unroll
        for (int k = 0; k < 4; ++k) {
            long long e = base + t + k * 256;
            if (e < E) {
                int idx = t + k * 256;
                int r = smr[b][idx];
                int c = smc[b][idx];
                if (alive[e] && (cur[r] | cur[c])) {
                    alive[e] = 0;
                    if (!last) { nxt[r] = 1; nxt[c] = 1; msk[r] = 1; msk[c] = 1; }
                }
            }
        }
        __syncthreads();
        b ^= 1;
    }
}

// Degree of surviving edges: deg[rows[e]] += 1 (L2-resident f32 atomics).
__global__ void deg_kernel(const int* __restrict__ rows,
                           const int* __restrict__ cols,
                           const int* __restrict__ alive,
                           float* __restrict__ deg, long long E) {
    __shared__ int smr[2][TILE_EDGES];
    __shared__ int smc[2][TILE_EDGES];
    int t = threadIdx.x;
    long long numTiles = (E + TILE_EDGES - 1) / TILE_EDGES;
    long long stride = gridDim.x;
    long long tile = blockIdx.x;
    int b = 0;
    if (tile < numTiles) issue_tile(rows, cols, tile * TILE_EDGES, E, smr, smc, b, t);
    for (; tile < numTiles; tile += stride) {
        wait_async0();
        __syncthreads();
        long long nt = tile + stride;
        if (nt < numTiles) issue_tile(rows, cols, nt * TILE_EDGES, E, smr, smc, b ^ 1, t);
        long long base = tile * TILE_EDGES;
#pragma unroll
        for (int k = 0; k < 4; ++k) {
            long long e = base + t + k * 256;
            if (e < E && alive[e]) {
                atomicAdd(&deg[smr[b][t + k * 256]], 1.0f);
            }
        }
        __syncthreads();
        b ^= 1;
    }
}

__global__ void dinv_kernel(const float* __restrict__ deg,
                            float* __restrict__ dinv, int n) {
    int i = blockIdx.x * blockDim.x + threadIdx.x;
    if (i < n) dinv[i] = rsqrtf(deg[i] + 1e-12f);
}

// enc_vals = alive ? dinv[r]*dinv[c] : 0; also emit rows/cols (as float).
__global__ void enc_kernel(const int* __restrict__ rows,
                           const int* __restrict__ cols,
                           const int* __restrict__ alive,
                           const float* __restrict__ dinv,
                           float* __restrict__ out,
                           long long E, long long osz) {
    __shared__ int smr[2][TILE_EDGES];
    __shared__ int smc[2][TILE_EDGES];
    int t = threadIdx.x;
    long long numTiles = (E + TILE_EDGES - 1) / TILE_EDGES;
    long long stride = gridDim.x;
    long long tile = blockIdx.x;
    int b = 0;
    if (tile < numTiles) issue_tile(rows, cols, tile * TILE_EDGES, E, smr, smc, b, t);
    for (; tile < numTiles; tile += stride) {
        wait_async0();
        __syncthreads();
        long long nt = tile + stride;
        if (nt < numTiles) issue_tile(rows, cols, nt * TILE_EDGES, E, smr, smc, b ^ 1, t);
        long long base = tile * TILE_EDGES;
#pragma unroll
        for (int k = 0; k < 4; ++k) {
            long long e = base + t + k * 256;
            if (e < E) {
                int idx = t + k * 256;
                int r = smr[b][idx];
                int c = smc[b][idx];
                float v = alive[e] ? dinv[r] * dinv[c] : 0.0f;
                if (e < osz)          out[e] = (float)r;
                if (E + e < osz)      out[E + e] = (float)c;
                if (2 * E + e < osz)  out[2 * E + e] = v;
            }
        }
        __syncthreads();
        b ^= 1;
    }
}

// ----------------------- deterministic mask compaction ---------------------
// Single block, 1024 threads = 32 wave32s; ballot + popcount per wave,
// serial scan of the 32 wave sums -> stable ascending-order compaction.
__global__ void compact_kernel(const int* __restrict__ msk,
                               int* __restrict__ midx,
                               int* __restrict__ mcnt, int n) {
    __shared__ int wsum[32];
    __shared__ int sbase;
    int t = threadIdx.x;
    if (t == 0) sbase = 0;
    __syncthreads();
    for (int base = 0; base < n; base += 1024) {
        int i = base + t;
        int f = (i < n) ? (msk[i] != 0) : 0;
        unsigned int b = (unsigned int)__ballot(f);   // wave32: low 32 bits
        int lane = t & 31;
        int wid = t >> 5;
        int pre = __popc(b & ((1u << lane) - 1u));
        if (lane == 0) wsum[wid] = __popc(b);
        __syncthreads();
        if (t == 0) {
            int acc = sbase;
            for (int w = 0; w < 32; ++w) { int v = wsum[w]; wsum[w] = acc; acc += v; }
            sbase = acc;
        }
        __syncthreads();
        if (f) midx[wsum[wid] + pre] = i;
        __syncthreads();
    }
    if (t == 0) mcnt[0] = sbase;
}

// ------------------------------ decoder fill -------------------------------
// Self loops first, then deterministic hash-random pairs among masked nodes.
__global__ void dec_fill_kernel(const int* __restrict__ midx,
                                const int* __restrict__ mcnt,
                                float* __restrict__ out,
                                long long base, long long cap, int n) {
    int cnt = *mcnt;
    if (cnt < 1) cnt = 1;
    long long stride = (long long)gridDim.x * blockDim.x;
    for (long long i = blockIdx.x * (long long)blockDim.x + threadIdx.x;
         i < cap; i += stride) {
        int r, c;
        if (i < n) {
            r = (int)i; c = (int)i;
        } else {
            unsigned u = (unsigned)i;
            r = midx[hashu(u * 2654435761u + 0x9E3779B9u) % (unsigned)cnt];
            c = midx[hashu(u * 40503u + 0x85EBCA6Bu) % (unsigned)cnt];
        }
        out[base + i] = (float)r;
        out[base + cap + i] = (float)c;
        out[base + 2 * cap + i] = 1.0f;
    }
}

// ------------------------------- launcher ----------------------------------

extern "C" void kernel_launch(void* const* d_in, const int* in_sizes, int n_in,
                              void* d_out, int out_size, void* d_ws, size_t ws_size,
                              hipStream_t stream) {
    const int* rows = (const int*)d_in[0];
    const int* cols = (const int*)d_in[1];
    // d_in[2] = vals (only its length matters in the reference; unused here)
    const int* seeds = (const int*)d_in[3];
    long long E = in_sizes[0];
    int S = in_sizes[3];
    const int N = 40000;        // N_USER + N_ITEM
    const int SAMP = 20000;     // N * KEEP_RATE
    float* out = (float*)d_out;
    long long osz = out_size;

    // workspace carve-out (16B aligned)
    char* w = (char*)d_ws;
    size_t off = 0;
    auto alloc = [&](size_t bytes) -> void* {
        void* p = w + off;
        off += (bytes + 15) & ~(size_t)15;
        return p;
    };
    int* cur   = (int*)alloc((size_t)N * 4);
    int* nxt   = (int*)alloc((size_t)N * 4);
    int* msk   = (int*)alloc((size_t)N * 4);
    int* midx  = (int*)alloc((size_t)N * 4);
    int* mcnt  = (int*)alloc(16);
    float* deg = (float*)alloc((size_t)N * 4);
    float* dnv = (float*)alloc((size_t)N * 4);
    int* alive = (int*)alloc((size_t)E * 4);
    if (off > ws_size) return;   // not enough scratch; bail deterministically

    dim3 blk(256);
    int nodeBlocks = (N + 255) / 256;
    const int sb = 640;          // persistent streaming blocks

    init_nodes_kernel<<<nodeBlocks, blk, 0, stream>>>(cur, nxt, msk, midx, deg, dnv, N);
    init_edges_kernel<<<1024, blk, 0, stream>>>(alive, E);
    if (osz > 0) zero_out_kernel<<<1024, blk, 0, stream>>>(out, osz);
    scatter_seeds_kernel<<<(S + 255) / 256, blk, 0, stream>>>(seeds, S, cur, msk, N);

    // MASK_DEPTH = 2: level 0 expands frontier, level 1 only kills edges.
    bfs_level_kernel<<<sb, blk, 0, stream>>>(rows, cols, alive, cur, nxt, msk, E, 0);
    bfs_level_kernel<<<sb, blk, 0, stream>>>(rows, cols, alive, nxt, cur, msk, E, 1);

    keep_sample_kernel<<<(SAMP + 255) / 256, blk, 0, stream>>>(msk, N, SAMP);

    deg_kernel<<<sb, blk, 0, stream>>>(rows, cols, alive, deg, E);
    dinv_kernel<<<nodeBlocks, blk, 0, stream>>>(deg, dnv, N);
    enc_kernel<<<sb, blk, 0, stream>>>(rows, cols, alive, dnv, out, E, osz);

    compact_kernel<<<1, 1024, 0, stream>>>(msk, midx, mcnt, N);

    long long cap = (osz > 3 * E) ? (osz - 3 * E) / 3 : 0;
    if (cap > 0)
        dec_fill_kernel<<<1024, blk, 0, stream>>>(midx, mcnt, out, 3 * E, cap, N);
}